// LightGCNTeacher_63763084477185
// MI455X (gfx1250) — compile-verified
//
#include <hip/hip_runtime.h>
#include <hip/hip_bf16.h>
#include <stdint.h>

// LightGCN propagation on MI455X (gfx1250, wave32).
// Memory/atomic-bound scatter SpMM; working set (~134 MB) fits the 192 MB L2.
// CDNA5-specific path used: GLOBAL_LOAD_ASYNC_TO_LDS_B32 (ASYNCcnt) for edge
// metadata staging + s_wait_asynccnt.

#define N_USERS 100000
#define N_ITEMS 50000
#define N_NODES (N_USERS + N_ITEMS)   // 150000
#define EMB 64
#define N_EDGES 4800000
#define EPB 256                        // edges staged per block
#define TOTAL_V4 (N_NODES * 16)        // float4 elements per node-matrix

// ---------------------------------------------------------------------------
// init: out = cur = concat(user_emb, item_emb); next = 0
// ---------------------------------------------------------------------------
__global__ void lgcn_init(const float4* __restrict__ ue,
                          const float4* __restrict__ ie,
                          float4* __restrict__ out,
                          float4* __restrict__ cur,
                          float4* __restrict__ nxt) {
    int i = blockIdx.x * blockDim.x + threadIdx.x;
    if (i >= TOTAL_V4) return;
    float4 v = (i < N_USERS * 16) ? ue[i] : ie[i - N_USERS * 16];
    out[i] = v;
    cur[i] = v;
    nxt[i] = make_float4(0.f, 0.f, 0.f, 0.f);
}

// ---------------------------------------------------------------------------
// spmm: next[row] += val * cur[col]   (COO scatter, f32 atomics)
// 16 lanes per edge, float4 per lane; edge metadata staged to LDS via the
// CDNA5 async-to-LDS path.
// ---------------------------------------------------------------------------
__global__ void lgcn_spmm(const int*   __restrict__ rows,
                          const int*   __restrict__ cols,
                          const float* __restrict__ vals,
                          const float4* __restrict__ cur,
                          float* __restrict__ nxt) {
    __shared__ int   s_row[EPB];
    __shared__ int   s_col[EPB];
    __shared__ float s_val[EPB];

    const int t    = threadIdx.x;               // 0..255
    const int base = blockIdx.x * EPB;          // N_EDGES % EPB == 0

    // ---- async stage of 256 edges' metadata into LDS (ASYNCcnt path) ----
    {
        int gbyte = (base + t) * 4;             // byte offset into each array
        unsigned lr = (unsigned)(uintptr_t)&s_row[t];
        unsigned lc = (unsigned)(uintptr_t)&s_col[t];
        unsigned lv = (unsigned)(uintptr_t)&s_val[t];
        asm volatile(
            "global_load_async_to_lds_b32 %0, %1, %6\n\t"
            "global_load_async_to_lds_b32 %2, %3, %7\n\t"
            "global_load_async_to_lds_b32 %4, %5, %8"
            :
            : "v"(lr), "v"(gbyte),
              "v"(lc), "v"(gbyte),
              "v"(lv), "v"(gbyte),
              "s"(rows), "s"(cols), "s"(vals)
            : "memory");
        asm volatile("s_wait_asynccnt 0" ::: "memory");
    }
    __syncthreads();

    const int g   = t & 15;      // float4 slot within the 64-wide embedding
    const int sub = t >> 4;      // which of 16 edges handled per pass

#pragma unroll 1
    for (int k = 0; k < EPB / 16; ++k) {
        const int   idx = k * 16 + sub;
        const int   row = s_row[idx];
        const int   col = s_col[idx];
        const float val = s_val[idx];

        // coalesced 256 B gather across the 16-lane group
        float4 v = cur[col * 16 + g];

        float* dst = nxt + (size_t)row * EMB + g * 4;
        atomicAdd(dst + 0, val * v.x);   // non-returning global_atomic_add_f32
        atomicAdd(dst + 1, val * v.y);
        atomicAdd(dst + 2, val * v.z);
        atomicAdd(dst + 3, val * v.w);
    }
}

// ---------------------------------------------------------------------------
// accumulate: out = (out + next) * scale ; if not last: cur = next, next = 0
// ---------------------------------------------------------------------------
__global__ void lgcn_accum(float4* __restrict__ out,
                           float4* __restrict__ cur,
                           float4* __restrict__ nxt,
                           float scale, int last) {
    int i = blockIdx.x * blockDim.x + threadIdx.x;
    if (i >= TOTAL_V4) return;
    float4 n = nxt[i];
    float4 o = out[i];
    o.x = (o.x + n.x) * scale;
    o.y = (o.y + n.y) * scale;
    o.z = (o.z + n.z) * scale;
    o.w = (o.w + n.w) * scale;
    out[i] = o;
    if (!last) {
        cur[i] = n;
        nxt[i] = make_float4(0.f, 0.f, 0.f, 0.f);
    }
}

// ---------------------------------------------------------------------------
extern "C" void kernel_launch(void* const* d_in, const int* in_sizes, int n_in,
                              void* d_out, int out_size, void* d_ws, size_t ws_size,
                              hipStream_t stream) {
    const float4* ue   = (const float4*)d_in[0];
    const float4* ie   = (const float4*)d_in[1];
    const int*    rows = (const int*)d_in[2];
    const int*    cols = (const int*)d_in[3];
    const float*  vals = (const float*)d_in[4];

    float4* out = (float4*)d_out;
    float*  cur = (float*)d_ws;                         // 38.4 MB
    float*  nxt = cur + (size_t)N_NODES * EMB;          // 38.4 MB

    const int ew_blocks = TOTAL_V4 / 256;               // 9375 (exact)
    const int sp_blocks = N_EDGES / EPB;                // 18750 (exact)

    lgcn_init<<<ew_blocks, 256, 0, stream>>>(ue, ie, out, (float4*)cur, (float4*)nxt);

    for (int layer = 0; layer < 3; ++layer) {
        lgcn_spmm<<<sp_blocks, 256, 0, stream>>>(rows, cols, vals,
                                                 (const float4*)cur, nxt);
        const int last = (layer == 2);
        lgcn_accum<<<ew_blocks, 256, 0, stream>>>(out, (float4*)cur, (float4*)nxt,
                                                  last ? 0.25f : 1.0f, last);
    }
}